// GNN_TransformerConv_21869973471855
// MI455X (gfx1250) — compile-verified
//
#include <hip/hip_runtime.h>
#include <hip/hip_bf16.h>
#include <cstdint>
#include <cstddef>

// ---------------------------------------------------------------------------
// Problem constants (from reference)
// ---------------------------------------------------------------------------
#define NN 50000      // nodes
#define NE 600000     // edges
#define HC 128        // H*C
#define HEADS 4
#define CH 32
#define EDIM 32
#define ODIM 64
#define MIDD 32
#define KMAX 128      // largest K used by any GEMM

typedef float v2f __attribute__((ext_vector_type(2)));
typedef float v8f __attribute__((ext_vector_type(8)));

enum { GF_BIAS = 1, GF_ACC = 2, GF_LEAKY = 4 };

__device__ __forceinline__ float leaky_f(float x) { return x > 0.0f ? x : 0.01f * x; }

// order-preserving float->uint encoding for atomicMax-based segment max
__device__ __forceinline__ unsigned encf(float x) {
  unsigned u = __float_as_uint(x);
  return (u & 0x80000000u) ? ~u : (u | 0x80000000u);
}
__device__ __forceinline__ float decf(unsigned e) {
  unsigned u = (e & 0x80000000u) ? (e & 0x7FFFFFFFu) : ~e;
  return __uint_as_float(u);
}

// ---------------------------------------------------------------------------
// Dense GEMM on the WMMA f32 pipe:  C[M x Nc] (op)= A[M x K] @ B[K x Nc]
//
// Workgroup = (Nc/16) waves; wave w owns the 16-column tile starting at 16*w.
// The block grid-strides over 16-row tiles of A. Each row tile (16 x K floats,
// <= 8 KB) is staged coalesced into LDS once and the fragments are then read
// via ds_load by all waves (8x reuse for Nc=128). The 16x16 accumulator runs
// as a V_WMMA_F32_16X16X4_F32 chain (exact f32, validation-safe).
// Requires M % 16 == 0, Nc % 16 == 0, K % 4 == 0, K <= KMAX.
// ---------------------------------------------------------------------------
__global__ __launch_bounds__(256)
void gemm_wmma_f32(const float* __restrict__ A, const float* __restrict__ B,
                   const float* __restrict__ bias, float* __restrict__ C,
                   int M, int K, int Nc, int flags)
{
  __shared__ float As[16 * KMAX];

  const int tid  = threadIdx.x;
  const int lane = tid & 31;
  const int wave = tid >> 5;        // column-tile index within block
  const int half = lane >> 4;       // 0: lanes 0-15, 1: lanes 16-31
  const int r    = lane & 15;
  const int col0 = wave * 16;
  const int nthreads = blockDim.x;
  const int mtiles = M >> 4;
  const int tile_elems = 16 * K;

  const float bv = (flags & GF_BIAS) ? bias[col0 + r] : 0.0f;
  const float* Bp = B + (size_t)(col0 + r);       // B-frag col = lane&15

  for (int rt = blockIdx.x; rt < mtiles; rt += gridDim.x) {
    const int row0 = rt * 16;

    // ---- stage 16 x K A-tile into LDS (coalesced row-major copy) ----
    __syncthreads();   // protect As from previous iteration's readers
    const float* Atile = A + (size_t)row0 * K;
    for (int idx = tid; idx < tile_elems; idx += nthreads)
      As[idx] = Atile[idx];
    __syncthreads();

    // ---- 16x16 WMMA accumulation over K ----
    v8f c = {};
    const float* Ap = &As[r * K];
    for (int k0 = 0; k0 < K; k0 += 4) {
      const int ka = k0 + 2 * half;               // per-lane K pair
      v2f a = *reinterpret_cast<const v2f*>(Ap + ka);   // ds_load_b64
      v2f b;
      b.x = Bp[(size_t)ka * Nc];                  // B[ka,   col] (hot in WGP$)
      b.y = Bp[(size_t)(ka + 1) * Nc];            // B[ka+1, col]
      c = __builtin_amdgcn_wmma_f32_16x16x4_f32(false, a, false, b,
                                                (short)0, c, false, false);
    }

    // ---- epilogue: bias / accumulate / LeakyReLU, coalesced store ----
#pragma unroll
    for (int i = 0; i < 8; ++i) {
      const int row = row0 + i + 8 * half;        // C VGPR i -> rows i / i+8
      const size_t idx = (size_t)row * Nc + col0 + r;
      float val = c[i] + bv;
      if (flags & GF_ACC)   val += C[idx];
      if (flags & GF_LEAKY) val = leaky_f(val);
      C[idx] = val;
    }
  }
}

// ---------------------------------------------------------------------------
// Utility fills
// ---------------------------------------------------------------------------
__global__ void fill_f32(float* __restrict__ p, float v, int n) {
  int i = blockIdx.x * blockDim.x + threadIdx.x;
  if (i < n) p[i] = v;
}
__global__ void fill_u32(unsigned* __restrict__ p, unsigned v, int n) {
  int i = blockIdx.x * blockDim.x + threadIdx.x;
  if (i < n) p[i] = v;
}

// Build block-diagonal 128x128 matrices from We (32x128):
//   BW[h*32+d, h*32+c] = We[d, h*32+c]     (for wea -> out transform)
//   BQ[h*32+c, h*32+d] = We[d, h*32+c]     (= BW^T, for q -> qwe transform)
__global__ void build_blockdiag(const float* __restrict__ We,
                                float* __restrict__ BW, float* __restrict__ BQ) {
  int idx = blockIdx.x * blockDim.x + threadIdx.x;   // 0 .. 16383
  if (idx >= HC * HC) return;
  int i = idx >> 7, j = idx & 127;
  bool same = (i >> 5) == (j >> 5);
  BW[idx] = same ? We[(size_t)(i & 31) * HC + j] : 0.0f;
  BQ[idx] = same ? We[(size_t)(j & 31) * HC + i] : 0.0f;
}

// ---------------------------------------------------------------------------
// Edge kernels
// ---------------------------------------------------------------------------
// alpha[e,h] = ( q[dst]_h . k[src]_h  +  qwe[dst]_h . ea[e] ) / sqrt(C)
// plus segment max via encoded atomicMax. One thread per (edge, head).
__global__ __launch_bounds__(256)
void edge_alpha(const int* __restrict__ ei, const float* __restrict__ q,
                const float* __restrict__ k, const float* __restrict__ qwe,
                const float* __restrict__ ea, float* __restrict__ alpha,
                unsigned* __restrict__ amaxEnc)
{
  int t = blockIdx.x * blockDim.x + threadIdx.x;
  int eid = t >> 2, hd = t & 3;
  if (eid >= NE) return;
  int s = ei[eid], d = ei[NE + eid];

  const float4* qp = reinterpret_cast<const float4*>(q   + (size_t)d * HC + hd * CH);
  const float4* kp = reinterpret_cast<const float4*>(k   + (size_t)s * HC + hd * CH);
  const float4* wp = reinterpret_cast<const float4*>(qwe + (size_t)d * HC + hd * CH);
  const float4* ep = reinterpret_cast<const float4*>(ea  + (size_t)eid * EDIM);

  float acc = 0.0f;
#pragma unroll
  for (int i = 0; i < CH / 4; ++i) {
    float4 a = qp[i], b = kp[i];
    acc = fmaf(a.x, b.x, acc); acc = fmaf(a.y, b.y, acc);
    acc = fmaf(a.z, b.z, acc); acc = fmaf(a.w, b.w, acc);
  }
#pragma unroll
  for (int i = 0; i < EDIM / 4; ++i) {
    float4 a = wp[i], b = ep[i];
    acc = fmaf(a.x, b.x, acc); acc = fmaf(a.y, b.y, acc);
    acc = fmaf(a.z, b.z, acc); acc = fmaf(a.w, b.w, acc);
  }
  acc *= 0.17677669529663687f;   // 1/sqrt(32)
  alpha[(size_t)eid * HEADS + hd] = acc;
  atomicMax(&amaxEnc[(size_t)d * HEADS + hd], encf(acc));
}

// a = exp(alpha - max[dst]); alpha <- a; asum[dst] += a
__global__ __launch_bounds__(256)
void edge_expsum(const int* __restrict__ ei, float* __restrict__ alpha,
                 const unsigned* __restrict__ amaxEnc, float* __restrict__ asum)
{
  int t = blockIdx.x * blockDim.x + threadIdx.x;
  int eid = t >> 2, hd = t & 3;
  if (eid >= NE) return;
  int d = ei[NE + eid];
  float m = decf(amaxEnc[(size_t)d * HEADS + hd]);
  float a = __expf(alpha[(size_t)eid * HEADS + hd] - m);
  alpha[(size_t)eid * HEADS + hd] = a;
  atomicAdd(&asum[(size_t)d * HEADS + hd], a);
}

// One thread per (edge, j=0..127):  h = j>>5
//   vacc[dst, j] += w * v[src, j]      (w = a / asum[dst,h])
//   wea [dst, j] += w * ea[e, j&31]
__global__ __launch_bounds__(256)
void edge_aggregate(const int* __restrict__ ei, const float* __restrict__ alpha,
                    const float* __restrict__ asum, const float* __restrict__ v,
                    const float* __restrict__ ea, float* __restrict__ vacc,
                    float* __restrict__ wea)
{
  int t = blockIdx.x * blockDim.x + threadIdx.x;   // < NE*128 = 76.8M
  int eid = t >> 7, j = t & 127;
  if (eid >= NE) return;
  int hd = j >> 5;
  int s = ei[eid], d = ei[NE + eid];
  float w = alpha[(size_t)eid * HEADS + hd] / asum[(size_t)d * HEADS + hd];
  atomicAdd(&vacc[(size_t)d * HC + j], w * v[(size_t)s * HC + j]);
  atomicAdd(&wea [(size_t)d * HC + j], w * ea[(size_t)eid * EDIM + (j & 31)]);
}

// H = leaky(out + vacc)
__global__ __launch_bounds__(256)
void add_act(const float* __restrict__ o, const float* __restrict__ vacc,
             float* __restrict__ H, int n)
{
  int i = blockIdx.x * blockDim.x + threadIdx.x;
  if (i < n) H[i] = leaky_f(o[i] + vacc[i]);
}

// ---------------------------------------------------------------------------
// Host-side helpers
// ---------------------------------------------------------------------------
static inline void launch_gemm(hipStream_t s, const float* A, const float* B,
                               const float* bias, float* C,
                               int M, int K, int Nc, int flags) {
  const int mtiles = M / 16;
  dim3 g(mtiles < 640 ? mtiles : 640), b(32 * (Nc / 16));
  gemm_wmma_f32<<<g, b, 0, s>>>(A, B, bias, C, M, K, Nc, flags);
}

static void run_layer(hipStream_t s, const float* X,
                      const float* Wq, const float* Wk, const float* Wv,
                      const float* Wsk, const float* We,
                      const float* bq, const float* bk, const float* bv,
                      const float* bsk, const int* ei, const float* ea,
                      float* bq_buf, float* bk_buf, float* bv_buf, float* bt_buf,
                      float* bo_buf, float* alpha, float* asum, unsigned* amaxE,
                      float* BW, float* BQ, float* outH)
{
  const int nb128 = NN * HC;
  // dense projections (WMMA)
  launch_gemm(s, X, Wq,  bq,  bq_buf, NN, HC, HC, GF_BIAS);
  launch_gemm(s, X, Wk,  bk,  bk_buf, NN, HC, HC, GF_BIAS);
  launch_gemm(s, X, Wv,  bv,  bv_buf, NN, HC, HC, GF_BIAS);
  launch_gemm(s, X, Wsk, bsk, bo_buf, NN, HC, HC, GF_BIAS);   // skip term
  // block-diagonal We matrices, then qwe = q @ BQ
  build_blockdiag<<<(HC * HC + 255) / 256, 256, 0, s>>>(We, BW, BQ);
  launch_gemm(s, bq_buf, BQ, nullptr, bt_buf, NN, HC, HC, 0);
  // segment softmax over incoming edges
  fill_u32<<<(NN * HEADS + 255) / 256, 256, 0, s>>>(amaxE, 0u, NN * HEADS);
  fill_f32<<<(NN * HEADS + 255) / 256, 256, 0, s>>>(asum, 0.0f, NN * HEADS);
  edge_alpha <<<(NE * HEADS + 255) / 256, 256, 0, s>>>(ei, bq_buf, bk_buf, bt_buf, ea, alpha, amaxE);
  edge_expsum<<<(NE * HEADS + 255) / 256, 256, 0, s>>>(ei, alpha, amaxE, asum);
  // weighted aggregation: vacc (reuses q buf), wea (reuses k buf)
  fill_f32<<<(nb128 + 255) / 256, 256, 0, s>>>(bq_buf, 0.0f, nb128);
  fill_f32<<<(nb128 + 255) / 256, 256, 0, s>>>(bk_buf, 0.0f, nb128);
  edge_aggregate<<<(NE * HC + 255) / 256, 256, 0, s>>>(ei, alpha, asum, bv_buf, ea, bq_buf, bk_buf);
  // out += wea @ BW  (WMMA, accumulate into skip), then H = leaky(out + vacc)
  launch_gemm(s, bk_buf, BW, nullptr, bo_buf, NN, HC, HC, GF_ACC);
  add_act<<<(nb128 + 255) / 256, 256, 0, s>>>(bo_buf, bq_buf, outH, nb128);
}

extern "C" void kernel_launch(void* const* d_in, const int* in_sizes, int n_in,
                              void* d_out, int out_size, void* d_ws, size_t ws_size,
                              hipStream_t stream)
{
  (void)in_sizes; (void)n_in; (void)out_size; (void)ws_size;
  const float* x  = (const float*)d_in[0];
  const int*   ei = (const int*)  d_in[2];   // [2, E] : src row, dst row
  const float* ea = (const float*)d_in[3];
  const float* Wq0  = (const float*)d_in[5],  *Wk0 = (const float*)d_in[6];
  const float* Wv0  = (const float*)d_in[7],  *Wsk0= (const float*)d_in[8];
  const float* We0  = (const float*)d_in[9];
  const float* bq0  = (const float*)d_in[10], *bk0 = (const float*)d_in[11];
  const float* bv0  = (const float*)d_in[12], *bsk0= (const float*)d_in[13];
  const float* Wq1  = (const float*)d_in[14], *Wk1 = (const float*)d_in[15];
  const float* Wv1  = (const float*)d_in[16], *Wsk1= (const float*)d_in[17];
  const float* We1  = (const float*)d_in[18];
  const float* bq1  = (const float*)d_in[19], *bk1 = (const float*)d_in[20];
  const float* bv1  = (const float*)d_in[21], *bsk1= (const float*)d_in[22];
  const float* W1 = (const float*)d_in[23], *b1 = (const float*)d_in[24];
  const float* W2 = (const float*)d_in[25], *b2 = (const float*)d_in[26];

  // Workspace carve-up (floats)
  float* base = (float*)d_ws;
  const size_t NB = (size_t)NN * HC;          // 6.4M floats
  float* bq_buf = base;                        // q   / later vacc
  float* bk_buf = base + 1 * NB;               // k   / later wea
  float* bv_buf = base + 2 * NB;               // v
  float* bt_buf = base + 3 * NB;               // qwe / layer-2 output
  float* bo_buf = base + 4 * NB;               // skip + wea@BW accumulator
  float* hbuf   = base + 5 * NB;               // layer-1 output
  size_t off    = 6 * NB;
  float* alpha  = base + off;  off += (size_t)NE * HEADS;
  float* asum   = base + off;  off += (size_t)NN * HEADS;
  unsigned* amaxE = (unsigned*)(base + off); off += (size_t)NN * HEADS;
  float* midb   = base + off;  off += (size_t)NN * MIDD;
  float* BW     = base + off;  off += (size_t)HC * HC;
  float* BQ     = base + off;  off += (size_t)HC * HC;

  // Layer 1: x -> hbuf
  run_layer(stream, x, Wq0, Wk0, Wv0, Wsk0, We0, bq0, bk0, bv0, bsk0,
            ei, ea, bq_buf, bk_buf, bv_buf, bt_buf, bo_buf,
            alpha, asum, amaxE, BW, BQ, hbuf);
  // Layer 2: hbuf -> bt_buf
  run_layer(stream, hbuf, Wq1, Wk1, Wv1, Wsk1, We1, bq1, bk1, bv1, bsk1,
            ei, ea, bq_buf, bk_buf, bv_buf, bt_buf, bo_buf,
            alpha, asum, amaxE, BW, BQ, bt_buf);
  // MLP head: mid = leaky(H2 @ W1 + b1); out = mid @ W2 + b2
  launch_gemm(stream, bt_buf, W1, b1, midb, NN, HC, MIDD, GF_BIAS | GF_LEAKY);
  launch_gemm(stream, midb, W2, b2, (float*)d_out, NN, MIDD, ODIM, GF_BIAS);
}